// YOLOTransform_8787503087895
// MI455X (gfx1250) — compile-verified
//
#include <hip/hip_runtime.h>
#include <stdint.h>

#define B_IMG  16
#define N_ANCH 25200
#define N_CH   85
#define N_CLS  80
#define TOPK   2048
#define MAXDET 300
#define CONF   0.25f
#define NMS_T  0.45f
#define MASK_W 64   // 2048 bits / 32

typedef float v2f __attribute__((ext_vector_type(2)));
typedef float v8f __attribute__((ext_vector_type(8)));

__device__ __forceinline__ uint32_t fkey(float f) {
  uint32_t u = __float_as_uint(f);
  return (u & 0x80000000u) ? ~u : (u | 0x80000000u);
}
__device__ __forceinline__ float unfkey(uint32_t k) {
  return __uint_as_float((k & 0x80000000u) ? (k ^ 0x80000000u) : ~k);
}

// ---------- Stage 1: masked score + argmax label per anchor (bandwidth-bound) ----
__global__ void score_label_kernel(const float* __restrict__ preds,
                                   float* __restrict__ scoresAll,
                                   int* __restrict__ labelsAll) {
  int i = blockIdx.x * blockDim.x + threadIdx.x;
  int b = blockIdx.y;
  if (i >= N_ANCH) return;
  const float* p = preds + ((size_t)b * N_ANCH + i) * N_CH;
  float sc = p[4];
  float best = p[5];
  int lab = 0;
#pragma unroll 4
  for (int c = 1; c < N_CLS; ++c) {
    float v = p[5 + c];
    if (v > best) { best = v; lab = c; }
  }
  scoresAll[(size_t)b * N_ANCH + i] = (sc > CONF) ? sc : -1.0f;
  labelsAll[(size_t)b * N_ANCH + i] = lab + 1;
}

// ---------- Stage 2: per-image radix-select top-2048 + bitonic sort + box decode --
__global__ void select_sort_kernel(const float* __restrict__ preds,
                                   const int* __restrict__ isz,
                                   const float* __restrict__ scoresAll,
                                   int* __restrict__ topIdx,
                                   float* __restrict__ topScore,
                                   float4* __restrict__ boxes) {
  __shared__ unsigned long long sbuf[TOPK];
  __shared__ uint32_t hist[256];
  __shared__ uint32_t s_prefix, s_above, s_hi, s_eq;
  const int b = blockIdx.x, tid = threadIdx.x, NT = blockDim.x;
  const float* sc = scoresAll + (size_t)b * N_ANCH;

  if (tid == 0) { s_prefix = 0u; s_above = 0u; }
  __syncthreads();

  // 4x 8-bit MSB-first radix select: find exact key of the 2048th largest.
  for (int pass = 0; pass < 4; ++pass) {
    int shift = 24 - 8 * pass;
    for (int j = tid; j < 256; j += NT) hist[j] = 0u;
    __syncthreads();
    uint32_t dmask = pass ? (0xFFFFFFFFu << (32 - 8 * pass)) : 0u;
    uint32_t sp = s_prefix;
    for (int i = tid; i < N_ANCH; i += NT) {
      uint32_t key = fkey(sc[i]);
      if ((key & dmask) == sp) atomicAdd(&hist[(key >> shift) & 0xFFu], 1u);
    }
    __syncthreads();
    if (tid == 0) {
      uint32_t need = TOPK - s_above, cum = 0u;
      int bin = 0;
      for (int bb = 255; bb >= 0; --bb) {
        uint32_t h = hist[bb];
        if (cum + h >= need) { bin = bb; break; }
        cum += h;
      }
      s_prefix = sp | ((uint32_t)bin << shift);
      s_above += cum;
    }
    __syncthreads();
  }
  const uint32_t pivot = s_prefix, nAbove = s_above, eqNeed = TOPK - nAbove;
  if (tid == 0) { s_hi = 0u; s_eq = 0u; }
  __syncthreads();

  // Compaction into LDS (order fixed by the sort below).
  for (int i = tid; i < N_ANCH; i += NT) {
    uint32_t key = fkey(sc[i]);
    unsigned long long pk =
        ((unsigned long long)key << 32) | (uint32_t)(~(uint32_t)i);
    if (key > pivot) {
      uint32_t p = atomicAdd(&s_hi, 1u);
      sbuf[p] = pk;
    } else if (key == pivot) {
      uint32_t e = atomicAdd(&s_eq, 1u);
      if (e < eqNeed) sbuf[nAbove + e] = pk;
    }
  }
  __syncthreads();

  // Bitonic sort, descending by (key, ~idx) -> matches jax top_k stable order.
  for (uint32_t kk = 2; kk <= TOPK; kk <<= 1) {
    for (uint32_t j = kk >> 1; j > 0; j >>= 1) {
      for (uint32_t t = tid; t < TOPK; t += NT) {
        uint32_t ix = t ^ j;
        if (ix > t) {
          unsigned long long A = sbuf[t], Bv = sbuf[ix];
          bool up = ((t & kk) == 0);
          if (up ? (A < Bv) : (A > Bv)) { sbuf[t] = Bv; sbuf[ix] = A; }
        }
      }
      __syncthreads();
    }
  }

  // Emit sorted scores/indices and decode + clip boxes.
  const float Hf = (float)isz[2 * b], Wf = (float)isz[2 * b + 1];
  const float rh = Hf / 640.0f, rw = Wf / 640.0f;
  for (int r = tid; r < TOPK; r += NT) {
    unsigned long long pk = sbuf[r];
    uint32_t key = (uint32_t)(pk >> 32);
    uint32_t idx = ~(uint32_t)pk;
    topScore[(size_t)b * TOPK + r] = unfkey(key);
    topIdx[(size_t)b * TOPK + r] = (int)idx;
    const float* pp = preds + ((size_t)b * N_ANCH + idx) * N_CH;
    float x = pp[0], y = pp[1], w = pp[2], h = pp[3];
    float xmin = x - 0.5f * w, ymin = y - 0.5f * h;
    float xmax = xmin + w, ymax = ymin + h;
    float4 bx;
    bx.x = fminf(fmaxf(xmin * rw, 0.f), Wf);
    bx.y = fminf(fmaxf(ymin * rh, 0.f), Hf);
    bx.z = fminf(fmaxf(xmax * rw, 0.f), Wf);
    bx.w = fminf(fmaxf(ymax * rh, 0.f), Hf);
    boxes[(size_t)b * TOPK + r] = bx;
  }
}

// ---------- Stage 3: suppression bitmask via WMMA rank-2 matmul --------------
// D[m][n] = area_i(m) + area_j(n) via V_WMMA_F32_16X16X4_F32 with
//   A row m = [area_i(m), 1, 0, 0] ; B col n = [1, area_j(n), 0, 0]^T
// One wave owns 16 rows; sweeps 64 iterations of 32 columns (2 WMMAs each).
// Threshold test uses inter > T*(union+eps) (union+eps > 0 always) -> no div.
__global__ void iou_mask_kernel(const float4* __restrict__ boxes,
                                uint32_t* __restrict__ mask) {
  __shared__ float4 ibx[16], jbx[16];
  __shared__ float areaI[16], areaJ[16];
  __shared__ uint32_t rowMaskT[MASK_W][16];  // [word][row], transposed
  const int b = blockIdx.x >> 7;      // /128
  const int i0 = (blockIdx.x & 127) << 4;
  const int lane = threadIdx.x;
  const float4* bx = boxes + (size_t)b * TOPK;

  if (lane < 16) {
    float4 v = bx[i0 + lane];
    ibx[lane] = v;
    areaI[lane] = fmaxf(v.z - v.x, 0.f) * fmaxf(v.w - v.y, 0.f);
  }
  const bool lo = lane < 16;
  const int lm = lane & 15;
  // A fragment (16x4 f32): VGPR0=K0/K2, VGPR1=K1/K3 per lane-half; K2,K3 = 0.
  v2f a;
  a.x = lo ? areaI[lm] : 0.f;
  a.y = lo ? 1.f : 0.f;

  for (int cw = 0; cw < TOPK / 32; ++cw) {   // 32 columns per iteration
    uint32_t wl[8], wh[8];                   // row r / row r+8 words (uniform)
#pragma unroll
    for (int half = 0; half < 2; ++half) {
      const int j0 = (cw << 5) + (half << 4);
      if (lane < 16) {
        float4 v = bx[j0 + lane];
        jbx[lane] = v;
        areaJ[lane] = fmaxf(v.z - v.x, 0.f) * fmaxf(v.w - v.y, 0.f);
      }
      // B fragment (4x16 f32): row K striped across lanes within a VGPR.
      v2f bf;
      bf.x = lo ? 1.f : 0.f;
      bf.y = lo ? areaJ[lm] : 0.f;
      v8f c = {0.f, 0.f, 0.f, 0.f, 0.f, 0.f, 0.f, 0.f};
      c = __builtin_amdgcn_wmma_f32_16x16x4_f32(false, a, false, bf, (short)0,
                                                c, false, false);
#pragma unroll
      for (int r = 0; r < 8; ++r) {
        const int m = r + (lo ? 0 : 8);  // C/D layout: VGPR r -> rows r / r+8
        const int n = lm;
        float4 bi = ibx[m], bj = jbx[n];
        float x1 = fmaxf(bi.x, bj.x), y1 = fmaxf(bi.y, bj.y);
        float x2 = fminf(bi.z, bj.z), y2 = fminf(bi.w, bj.w);
        float inter = fmaxf(x2 - x1, 0.f) * fmaxf(y2 - y1, 0.f);
        float uni = c[r] - inter;        // area_i + area_j from WMMA
        // iou > T  <=>  inter > T*(uni + eps)   (uni + eps > 0)
        bool pred = (inter > NMS_T * (uni + 1e-7f)) && ((i0 + m) != (j0 + n));
        uint32_t bal = __builtin_amdgcn_ballot_w32(pred);
        if (half == 0) {
          wl[r] = bal & 0xFFFFu;
          wh[r] = bal >> 16;
        } else {
          wl[r] |= (bal & 0xFFFFu) << 16;
          wh[r] |= (bal >> 16) << 16;
        }
      }
    }
    if (lane == 0) {  // 16 contiguous words -> 4x ds_store_b128
      uint4* dst = (uint4*)&rowMaskT[cw][0];
      dst[0] = make_uint4(wl[0], wl[1], wl[2], wl[3]);
      dst[1] = make_uint4(wl[4], wl[5], wl[6], wl[7]);
      dst[2] = make_uint4(wh[0], wh[1], wh[2], wh[3]);
      dst[3] = make_uint4(wh[4], wh[5], wh[6], wh[7]);
    }
  }
  uint32_t* mrow = mask + ((size_t)b * TOPK + i0) * MASK_W;
  for (int t = lane; t < 16 * MASK_W; t += 32)
    mrow[t] = rowMaskT[t & 63][t >> 6];
}

// ---------- Stage 4: serial greedy scan over bitmask (early exit @300) -------
__global__ void nms_scan_kernel(const float* __restrict__ topScore,
                                const uint32_t* __restrict__ mask,
                                int* __restrict__ keepList,
                                int* __restrict__ keepCount) {
  __shared__ uint32_t removed[MASK_W];
  __shared__ int cnt;
  const int b = blockIdx.x, tid = threadIdx.x;  // blockDim = 64
  removed[tid] = 0u;
  if (tid == 0) cnt = 0;
  __syncthreads();
  for (int i = 0; i < TOPK; ++i) {
    bool sup = (removed[i >> 5] >> (i & 31)) & 1u;
    float sc = topScore[(size_t)b * TOPK + i];
    bool take = (!sup) && (sc > CONF);
    if (take) {
      if (tid == 0) {
        if (cnt < MAXDET) keepList[b * MAXDET + cnt] = i;
        cnt = cnt + 1;
      }
      removed[tid] |= mask[((size_t)b * TOPK + i) * MASK_W + tid];
    }
    __syncthreads();
    if (cnt >= MAXDET) break;
    __syncthreads();
  }
  if (tid == 0) keepCount[b] = (cnt < MAXDET) ? cnt : MAXDET;
}

// ---------- Stage 5: emit [16][300][6] ---------------------------------------
__global__ void output_kernel(const float4* __restrict__ boxes,
                              const float* __restrict__ topScore,
                              const int* __restrict__ topIdx,
                              const int* __restrict__ labelsAll,
                              const int* __restrict__ keepList,
                              const int* __restrict__ keepCount,
                              float* __restrict__ out) {
  const int b = blockIdx.x, r = threadIdx.x;
  if (r >= MAXDET) return;
  float o[6] = {0.f, 0.f, 0.f, 0.f, 0.f, 0.f};
  int cnt = keepCount[b];
  if (r < cnt) {
    int rank = keepList[b * MAXDET + r];
    float4 bx = boxes[(size_t)b * TOPK + rank];
    float sc = topScore[(size_t)b * TOPK + rank];
    int idx = topIdx[(size_t)b * TOPK + rank];
    int lab = labelsAll[(size_t)b * N_ANCH + idx];
    o[0] = bx.x; o[1] = bx.y; o[2] = bx.z; o[3] = bx.w;
    o[4] = sc;   o[5] = (float)lab;
  }
  float* po = out + ((size_t)b * MAXDET + r) * 6;
  for (int q = 0; q < 6; ++q) po[q] = o[q];
}

extern "C" void kernel_launch(void* const* d_in, const int* in_sizes, int n_in,
                              void* d_out, int out_size, void* d_ws,
                              size_t ws_size, hipStream_t stream) {
  const float* preds = (const float*)d_in[0];
  const int* isz = (const int*)d_in[1];
  char* ws = (char*)d_ws;
  size_t off = 0;
  float* scoresAll = (float*)(ws + off); off += (size_t)B_IMG * N_ANCH * 4;
  int* labelsAll   = (int*)(ws + off);   off += (size_t)B_IMG * N_ANCH * 4;
  int* topIdx      = (int*)(ws + off);   off += (size_t)B_IMG * TOPK * 4;
  float* topScore  = (float*)(ws + off); off += (size_t)B_IMG * TOPK * 4;
  float4* boxes    = (float4*)(ws + off); off += (size_t)B_IMG * TOPK * 16;
  uint32_t* mask   = (uint32_t*)(ws + off); off += (size_t)B_IMG * TOPK * MASK_W * 4;
  int* keepList    = (int*)(ws + off);   off += (size_t)B_IMG * MAXDET * 4;
  int* keepCount   = (int*)(ws + off);   off += (size_t)B_IMG * 4;

  dim3 g1((N_ANCH + 255) / 256, B_IMG);
  score_label_kernel<<<g1, 256, 0, stream>>>(preds, scoresAll, labelsAll);
  select_sort_kernel<<<B_IMG, 1024, 0, stream>>>(preds, isz, scoresAll, topIdx,
                                                 topScore, boxes);
  iou_mask_kernel<<<B_IMG * 128, 32, 0, stream>>>(boxes, mask);
  nms_scan_kernel<<<B_IMG, 64, 0, stream>>>(topScore, mask, keepList, keepCount);
  output_kernel<<<B_IMG, 320, 0, stream>>>(boxes, topScore, topIdx, labelsAll,
                                           keepList, keepCount, (float*)d_out);
}